// BA_Shapes_GCN_50646254354931
// MI455X (gfx1250) — compile-verified
//
#include <hip/hip_runtime.h>

#define NEG_SLOPE 0.01f

typedef __attribute__((ext_vector_type(2))) float v2f;
typedef __attribute__((ext_vector_type(8))) float v8f;

// ---------------- degree / normalization ----------------

__global__ void k_init_deg(float* __restrict__ deg, int N) {
  int i = blockIdx.x * blockDim.x + threadIdx.x;
  if (i < N) deg[i] = 1.0f;  // self-loop contributes 1 to every node's degree
}

__global__ void k_count_deg(const int* __restrict__ dst, float* __restrict__ deg, int E) {
  int e = blockIdx.x * blockDim.x + threadIdx.x;
  if (e < E) atomicAdd(&deg[dst[e]], 1.0f);
}

__global__ void k_dinv(float* __restrict__ deg, int N) {
  int i = blockIdx.x * blockDim.x + threadIdx.x;
  if (i < N) deg[i] = rsqrtf(deg[i]);  // deg >= 1 always
}

// ---------------- WMMA fp32 GEMM: out[N,Hout] = A[N,K] @ W[K,Hout] ----------------
// 256-thread block = 8 wave32s. The block owns a 128-row x 16-col output panel:
// wave w computes the 16x16 tile at rows blockIdx.x*128 + w*16, cols blockIdx.y*16.
// The K x 16 panel of W is staged in LDS once per block and shared by all 8 waves
// (ds_load_b32 reads; 8x fewer global B loads).
//
// V_WMMA_F32_16X16X4_F32 operand layout (ISA 7.12.2):
//   A 16x4: lanes 0-15 -> K=0(.x),1(.y); lanes 16-31 -> K=2(.x),3(.y)
//   B  4x16: symmetric, row striped across lanes
//   C/D v8f: lanes 0-15 -> M=r; lanes 16-31 -> M=r+8; N=lane%16
__global__ void __launch_bounds__(256)
k_gemm_wmma(const float* __restrict__ A, const float* __restrict__ W,
            float* __restrict__ out, int N, int K, int Hout) {
  __shared__ float sW[128 * 16];  // max K=128; 8 KB of the 320 KB WGP LDS

  const int tid  = threadIdx.x;
  const int wave = tid >> 5;
  const int lane = tid & 31;
  const int half = lane >> 4;   // 0 or 1
  const int lid  = lane & 15;
  const int col0 = blockIdx.y * 16;
  const int row0 = blockIdx.x * 128 + wave * 16;

  // cooperative stage: sW[k*16 + n] = W[k*Hout + col0 + n]
  for (int idx = tid; idx < K * 16; idx += 256) {
    int k = idx >> 4;
    int n = idx & 15;
    sW[idx] = W[(size_t)k * Hout + col0 + n];
  }
  __syncthreads();

  int m  = row0 + lid;
  int mc = (m < N) ? m : (N - 1);  // clamp loads; EXEC stays all-ones for WMMA
  const float* arow = A + (size_t)mc * K;

  v8f acc = {};
  for (int k = 0; k < K; k += 4) {
    const int ka = k + 2 * half;
    v2f a = *(const v2f*)(arow + ka);             // global_load_b64
    v2f b;
    b.x = sW[ka * 16 + lid];                      // ds_load_b32
    b.y = sW[(ka + 1) * 16 + lid];
    // 8 args: (neg_a, A, neg_b, B, c_mod, C, reuse_a, reuse_b)
    acc = __builtin_amdgcn_wmma_f32_16x16x4_f32(false, a, false, b, (short)0, acc,
                                                false, false);
  }

#pragma unroll
  for (int r = 0; r < 8; ++r) {
    int mr = row0 + r + 8 * half;
    if (mr < N) out[(size_t)mr * Hout + col0 + lid] = acc[r];
  }
}

// ---------------- aggregation ----------------

// Self-loop term: agg[i,f] = h[i,f] * dinv[i]^2   (initializes agg buffer)
__global__ void k_selfloop(const float* __restrict__ h, const float* __restrict__ dinv,
                           float* __restrict__ agg, int N, int H) {
  size_t idx = (size_t)blockIdx.x * blockDim.x + threadIdx.x;
  size_t tot = (size_t)N * H;
  if (idx >= tot) return;
  int i = (int)(idx / H);
  float di = dinv[i];
  agg[idx] = h[idx] * di * di;
}

// Edge scatter-add: agg[dst,f] += h[src,f] * dinv[src]*dinv[dst]
// Consecutive threads = consecutive features of one edge -> coalesced gather + atomics.
// h (12.8 MB) + edge lists (6.4 MB) are resident in the 192 MB L2.
__global__ void k_edge(const float* __restrict__ h, const float* __restrict__ dinv,
                       const int* __restrict__ src, const int* __restrict__ dst,
                       float* __restrict__ agg, int E, int H) {
  size_t idx = (size_t)blockIdx.x * blockDim.x + threadIdx.x;
  size_t tot = (size_t)E * H;
  if (idx >= tot) return;
  int e = (int)(idx / H);
  int f = (int)(idx % H);
  int s = src[e];
  int d = dst[e];
  float nrm = dinv[s] * dinv[d];
  atomicAdd(&agg[(size_t)d * H + f], h[(size_t)s * H + f] * nrm);
}

// act[i,f] = leaky_relu(agg[i,f] + b[f])
__global__ void k_bias_act(const float* __restrict__ agg, const float* __restrict__ b,
                           float* __restrict__ act, int N, int H) {
  size_t idx = (size_t)blockIdx.x * blockDim.x + threadIdx.x;
  size_t tot = (size_t)N * H;
  if (idx >= tot) return;
  int f = (int)(idx % H);
  float v = agg[idx] + b[f];
  act[idx] = (v > 0.0f) ? v : NEG_SLOPE * v;
}

// ---------------- head: concepts = exp(h - rowmax) ; logits = concepts @ Wl + bl ----
__global__ void k_head(const float* __restrict__ h3, const float* __restrict__ Wl,
                       const float* __restrict__ bl, float* __restrict__ out, int N) {
  const int C = 32, K = 4;
  int i = blockIdx.x * blockDim.x + threadIdx.x;
  if (i >= N) return;

  float v[C];
  float mx = -3.402823466e+38f;
#pragma unroll
  for (int j = 0; j < C; ++j) {
    v[j] = h3[(size_t)i * C + j];
    mx = fmaxf(mx, v[j]);
  }

  float lg[K];
#pragma unroll
  for (int k = 0; k < K; ++k) lg[k] = bl[k];

#pragma unroll
  for (int j = 0; j < C; ++j) {
    // softmax(p)/max(p) == exp(h_j - max): the softmax normalizers cancel
    float c = __expf(v[j] - mx);
    out[(size_t)i * C + j] = c;
#pragma unroll
    for (int k = 0; k < K; ++k) lg[k] += c * Wl[j * K + k];
  }

  float* lout = out + (size_t)N * C;
#pragma unroll
  for (int k = 0; k < K; ++k) lout[(size_t)i * K + k] = lg[k];
}

// ---------------- launch ----------------

extern "C" void kernel_launch(void* const* d_in, const int* in_sizes, int n_in,
                              void* d_out, int out_size, void* d_ws, size_t ws_size,
                              hipStream_t stream) {
  const float* x   = (const float*)d_in[0];
  const int*   src = (const int*)  d_in[1];
  const int*   dst = (const int*)  d_in[2];
  const float* W0  = (const float*)d_in[3];
  const float* b0  = (const float*)d_in[4];
  const float* W1  = (const float*)d_in[5];
  const float* b1  = (const float*)d_in[6];
  const float* W2  = (const float*)d_in[7];
  const float* b2  = (const float*)d_in[8];
  const float* W3  = (const float*)d_in[9];
  const float* b3  = (const float*)d_in[10];
  const float* Wl  = (const float*)d_in[11];
  const float* bl  = (const float*)d_in[12];

  const int F_in = 128, H = 64, C = 32;
  const int E = in_sizes[1];
  const int N = in_sizes[0] / F_in;

  // workspace layout (floats): dinv[N] | hbuf[N*64] | aggbuf[N*64] | actbuf[N*64]
  float* ws     = (float*)d_ws;
  float* dinv   = ws;
  float* hbuf   = dinv + N;
  float* aggbuf = hbuf + (size_t)N * H;
  float* actbuf = aggbuf + (size_t)N * H;

  const int T = 256;
  k_init_deg<<<(N + T - 1) / T, T, 0, stream>>>(dinv, N);
  k_count_deg<<<(E + T - 1) / T, T, 0, stream>>>(dst, dinv, E);
  k_dinv<<<(N + T - 1) / T, T, 0, stream>>>(dinv, N);

  const float* Ws[4] = {W0, W1, W2, W3};
  const float* bs[4] = {b0, b1, b2, b3};
  const int Kd[4] = {F_in, H, H, H};
  const int Ho[4] = {H, H, H, C};

  const float* in = x;
  for (int l = 0; l < 4; ++l) {
    dim3 grid((N + 127) / 128, Ho[l] / 16);
    k_gemm_wmma<<<grid, 256, 0, stream>>>(in, Ws[l], hbuf, N, Kd[l], Ho[l]);

    size_t ntot = (size_t)N * Ho[l];
    k_selfloop<<<(unsigned)((ntot + T - 1) / T), T, 0, stream>>>(hbuf, dinv, aggbuf, N, Ho[l]);

    size_t etot = (size_t)E * Ho[l];
    k_edge<<<(unsigned)((etot + T - 1) / T), T, 0, stream>>>(hbuf, dinv, src, dst, aggbuf, E, Ho[l]);

    k_bias_act<<<(unsigned)((ntot + T - 1) / T), T, 0, stream>>>(aggbuf, bs[l], actbuf, N, Ho[l]);
    in = actbuf;
  }

  k_head<<<(N + T - 1) / T, T, 0, stream>>>(actbuf, Wl, bl, (float*)d_out, N);
}